// DTCWTForward_old_30253749633559
// MI455X (gfx1250) — compile-verified
//
#include <hip/hip_runtime.h>
#include <hip/hip_bf16.h>

// DTCWT forward, MI455X (gfx1250), fp32.
// Level 1 (biort 5/7-tap) implemented as banded matmuls on V_WMMA_F32_16X16X4_F32.
// Levels 2-3 (qshift 14-tap, lowpass-only since highs are discarded) as VALU
// separable stride-2 convs with LDS staging.
//
// Workspace layout (d_ws, floats):
//   lolo1: 256 images * 256*256           = 16,777,216 floats (67.1 MB)
//   lolo2: 256 images * 128*128           =  4,194,304 floats (16.8 MB)
//   total ~84 MB, fully rewritten every call.

typedef __attribute__((ext_vector_type(2))) float v2f;
typedef __attribute__((ext_vector_type(8))) float v8f;

__device__ __forceinline__ int refl(int p, int n) {
  // symmetric half-sample reflection: -1->0, -2->1, n->n-1, n+1->n-2 ...
  if (p < 0) p = -p - 1;
  if (p >= n) p = 2 * n - 1 - p;
  return p;
}

// banded filter value: contribution of (slab index k) to output index n,
// slab aligned so k=0 <-> output-base - 3 (the 7-tap halo).
__device__ __forceinline__ float band7(const float* h, int d) { // d = k - n
  return (d >= 0 && d <= 6) ? h[6 - d] : 0.0f;
}
__device__ __forceinline__ float band5(const float* h, int d) {
  return (d >= 1 && d <= 5) ? h[5 - d] : 0.0f;
}

// ---------------------------------------------------------------------------
// Level 1: biort level. One workgroup = 32x64 output tile of one 256x256 image.
// ---------------------------------------------------------------------------
#define L1_TR 32
#define L1_TC 64
#define MIDR  48   // 3 WMMA row-blocks of Mid (needs 38, rounded up)
#define INW   72   // input slab col stride (70 needed + pad; all 72 loaded valid)
#define MIDW  64

__global__ __launch_bounds__(256)
void dtcwt_level1_wmma(const float* __restrict__ x,
                       const float* __restrict__ h0o,
                       const float* __restrict__ h1o,
                       float* __restrict__ lolo,   // 256 x (256*256)
                       float* __restrict__ yhr,    // 256 x 6 x 128 x 128
                       float* __restrict__ yhi)
{
  __shared__ float s_in[MIDR * INW];        // 48x72 input slab
  __shared__ float s_midLo[MIDR * MIDW];    // horizontal lowpass
  __shared__ float s_midHi[MIDR * MIDW];    // horizontal highpass
  __shared__ float s_hp[3][L1_TR * L1_TC];  // LoHi, HiLo, HiHi tiles

  const int img = blockIdx.z;            // b*64 + c
  const int r0  = blockIdx.y * L1_TR;
  const int c0  = blockIdx.x * L1_TC;
  const float* xin = x + (size_t)img * 256 * 256;

  const int tid  = threadIdx.x;
  const int lane = tid & 31;
  const int wave = tid >> 5;             // 8 waves
  const int half = lane >> 4;            // lane half selects K pair
  const int ln   = lane & 15;            // M (A-frag) or N (B-frag)

  float f0[5], f1[7];
#pragma unroll
  for (int i = 0; i < 5; ++i) f0[i] = h0o[i];
#pragma unroll
  for (int i = 0; i < 7; ++i) f1[i] = h1o[i];

  // Banded-filter WMMA fragments. For F32 16x16x4:
  //   A-frag: lane holds M=ln, VGPR{0,1} = K = 4s+2*half + {0,1}
  //   B-frag: lane holds N=ln, VGPR{0,1} = K rows 4s+2*half + {0,1}
  // Band matrices are symmetric in role: G[m][k] = B[k][n] = band(k - index),
  // so one fragment set serves horizontal-B and vertical-A.
  v2f fragLo[6], fragHi[6];
#pragma unroll
  for (int s = 0; s < 6; ++s) {
    int k = 4 * s + 2 * half;
    fragLo[s].x = band5(f0, k - ln);
    fragLo[s].y = band5(f0, k + 1 - ln);
    fragHi[s].x = band7(f1, k - ln);
    fragHi[s].y = band7(f1, k + 1 - ln);
  }

  // -------- Phase 1: load input slab (48 rows x 72 cols), reflected --------
  for (int idx = tid; idx < MIDR * INW; idx += 256) {
    int j = idx / INW, k = idx % INW;
    int rr = refl(r0 - 3 + j, 256);
    int cc = refl(c0 - 3 + k, 256);
    s_in[j * INW + k] = xin[rr * 256 + cc];
  }
  __syncthreads();

  // -------- Phase 2: horizontal conv via WMMA: Mid = X(16x24) * F(24x16) ----
  // 24 jobs = {Lo,Hi} x 3 row-blocks x 4 col-blocks
  for (int job = wave; job < 24; job += 8) {
    int f  = job % 2;          // 0=Lo(h0o), 1=Hi(h1o)
    int rb = (job / 2) % 3;
    int cb = job / 6;
    const int rowbase = rb * 16;
    const int cbase   = cb * 16;

    v8f acc = {0.f, 0.f, 0.f, 0.f, 0.f, 0.f, 0.f, 0.f};
#pragma unroll
    for (int s = 0; s < 6; ++s) {
      int k = 4 * s + 2 * half;
      v2f a;
      a.x = s_in[(rowbase + ln) * INW + cbase + k];
      a.y = s_in[(rowbase + ln) * INW + cbase + k + 1];
      v2f b = f ? fragHi[s] : fragLo[s];
      acc = __builtin_amdgcn_wmma_f32_16x16x4_f32(
          false, a, false, b, (short)0, acc, false, false);
    }
    float* smid = f ? s_midHi : s_midLo;
#pragma unroll
    for (int v = 0; v < 8; ++v)   // D: row = v + 8*half, col = ln
      smid[(rowbase + half * 8 + v) * MIDW + cbase + ln] = acc[v];
  }
  __syncthreads();

  // -------- Phase 3: vertical conv via WMMA: Out = G(16x24) * Mid(24x16) ---
  // 32 jobs = 4 combos x 2 out-row-blocks x 4 col-blocks
  for (int job = wave; job < 32; job += 8) {
    int combo = job & 3;          // 0=LoLo 1=LoHi 2=HiLo 3=HiHi
    int orb   = (job >> 2) & 1;
    int cb    = job >> 3;
    const float* smid = (combo <= 1) ? s_midLo : s_midHi;
    const bool hiv = (combo == 1) || (combo == 3);   // vertical filter h1o?
    const int kbase = orb * 16;   // Mid slab row base (slab row 0 <-> r0-3)

    v8f acc = {0.f, 0.f, 0.f, 0.f, 0.f, 0.f, 0.f, 0.f};
#pragma unroll
    for (int s = 0; s < 6; ++s) {
      int k = 4 * s + 2 * half;
      v2f a = hiv ? fragHi[s] : fragLo[s];   // G[m][k] = band(k - m)
      v2f b;
      b.x = smid[(kbase + k)     * MIDW + cb * 16 + ln];
      b.y = smid[(kbase + k + 1) * MIDW + cb * 16 + ln];
      acc = __builtin_amdgcn_wmma_f32_16x16x4_f32(
          false, a, false, b, (short)0, acc, false, false);
    }

    if (combo == 0) {             // LoLo -> workspace image
      float* dst = lolo + (size_t)img * 65536;
#pragma unroll
      for (int v = 0; v < 8; ++v) {
        int R = r0 + orb * 16 + half * 8 + v;
        int Cc = c0 + cb * 16 + ln;
        dst[R * 256 + Cc] = acc[v];
      }
    } else {                      // highpasses -> LDS for q2c
      float* sh = s_hp[combo - 1];
#pragma unroll
      for (int v = 0; v < 8; ++v)
        sh[(orb * 16 + half * 8 + v) * L1_TC + cb * 16 + ln] = acc[v];
    }
  }
  __syncthreads();

  // -------- Phase 4: q2c + store Yhr/Yhi -----------------------------------
  // s_hp[0]=LoHi -> orients (0,5); s_hp[1]=HiLo -> (2,3); s_hp[2]=HiHi -> (1,4)
  const float SC = 0.70710678118654752f;
  const int i0 = r0 >> 1, j0 = c0 >> 1;
  for (int idx = tid; idx < 3 * 16 * 32; idx += 256) {
    int srcn = idx / 512;
    int rem  = idx % 512;
    int i = rem >> 5, j = rem & 31;
    const float* sh = s_hp[srcn];
    float a = sh[(2 * i) * L1_TC + 2 * j];
    float b = sh[(2 * i) * L1_TC + 2 * j + 1];
    float c = sh[(2 * i + 1) * L1_TC + 2 * j];
    float d = sh[(2 * i + 1) * L1_TC + 2 * j + 1];
    int o1, o2;
    if (srcn == 0)      { o1 = 0; o2 = 5; }
    else if (srcn == 1) { o1 = 2; o2 = 3; }
    else                { o1 = 1; o2 = 4; }
    size_t p1 = ((size_t)img * 6 + o1) * 16384 + (size_t)(i0 + i) * 128 + (j0 + j);
    size_t p2 = ((size_t)img * 6 + o2) * 16384 + (size_t)(i0 + i) * 128 + (j0 + j);
    yhr[p1] = SC * (a - d);
    yhi[p1] = SC * (b + c);
    yhr[p2] = SC * (a + d);
    yhi[p2] = SC * (b - c);
  }
}

// ---------------------------------------------------------------------------
// Levels 2-3: qshift lowpass-only, separable stride-2 14-tap, fused via LDS.
// out[2i]   = sum_t h0b[t] * x[refl(4i+14-2t)]
// out[2i+1] = sum_t h0a[t] * x[refl(4i+15-2t)]
// One workgroup = 16 rows x 64 cols of the (L/2)x(L/2) output.
// ---------------------------------------------------------------------------
#define QS_INW  153   // 152 used (odd stride -> no LDS bank conflicts)
#define QS_MIDW 65    // 64 used

__global__ __launch_bounds__(256)
void dtcwt_qshift_lo(const float* __restrict__ src, float* __restrict__ dst,
                     const float* __restrict__ h0a, const float* __restrict__ h0b,
                     int L, int OS, int dstStride)
{
  __shared__ float s_in[56 * QS_INW];
  __shared__ float s_mid[56 * QS_MIDW];

  const int img = blockIdx.z;
  const int o0  = blockIdx.y * 16;   // output row base
  const int oc0 = blockIdx.x * 64;   // output col base
  const float* xin = src + (size_t)img * L * L;
  const int tid = threadIdx.x;

  float fa[14], fb[14];
#pragma unroll
  for (int i = 0; i < 14; ++i) { fa[i] = h0a[i]; fb[i] = h0b[i]; }

  // input slab: rows abs = refl(2*o0-12+jr), cols abs = refl(2*oc0-12+kc)
  for (int idx = tid; idx < 56 * 152; idx += 256) {
    int jr = idx / 152, kc = idx % 152;
    int rr = refl(2 * o0 - 12 + jr, L);
    int cc = refl(2 * oc0 - 12 + kc, L);
    s_in[jr * QS_INW + kc] = xin[(size_t)rr * L + cc];
  }
  __syncthreads();

  // horizontal pass: Mid[jr][ocr], slab col idx = 4u + (26|27) - 2t
  for (int idx = tid; idx < 56 * 64; idx += 256) {
    int jr = idx / 64, ocr = idx % 64;
    int u = ocr >> 1;
    const float* row = s_in + jr * QS_INW;
    float acc = 0.0f;
    if ((ocr & 1) == 0) {
#pragma unroll
      for (int t = 0; t < 14; ++t) acc = fmaf(fb[t], row[4 * u + 26 - 2 * t], acc);
    } else {
#pragma unroll
      for (int t = 0; t < 14; ++t) acc = fmaf(fa[t], row[4 * u + 27 - 2 * t], acc);
    }
    s_mid[jr * QS_MIDW + ocr] = acc;
  }
  __syncthreads();

  // vertical pass: out[ir][ocr], slab row idx = 4u + (26|27) - 2t
  for (int idx = tid; idx < 16 * 64; idx += 256) {
    int ir = idx / 64, ocr = idx % 64;
    int u = ir >> 1;
    float acc = 0.0f;
    if ((ir & 1) == 0) {
#pragma unroll
      for (int t = 0; t < 14; ++t)
        acc = fmaf(fb[t], s_mid[(4 * u + 26 - 2 * t) * QS_MIDW + ocr], acc);
    } else {
#pragma unroll
      for (int t = 0; t < 14; ++t)
        acc = fmaf(fa[t], s_mid[(4 * u + 27 - 2 * t) * QS_MIDW + ocr], acc);
    }
    dst[(size_t)img * dstStride + (size_t)(o0 + ir) * OS + (oc0 + ocr)] = acc;
  }
}

// ---------------------------------------------------------------------------
extern "C" void kernel_launch(void* const* d_in, const int* in_sizes, int n_in,
                              void* d_out, int out_size, void* d_ws, size_t ws_size,
                              hipStream_t stream) {
  (void)in_sizes; (void)n_in; (void)out_size; (void)ws_size;
  const float* x   = (const float*)d_in[0];
  const float* h0o = (const float*)d_in[1];
  const float* h1o = (const float*)d_in[2];
  const float* h0a = (const float*)d_in[3];
  const float* h0b = (const float*)d_in[4];
  // d_in[5]=h1a, d_in[6]=h1b unused: qshift highpasses are discarded.

  float* out = (float*)d_out;
  float* yl  = out;                        // 4*64*64*64      = 1,048,576
  float* yhr = out + 1048576;              // 4*64*6*128*128  = 25,165,824
  float* yhi = yhr + 25165824;

  float* lolo1 = (float*)d_ws;             // 16,777,216 floats
  float* lolo2 = lolo1 + 16777216;         //  4,194,304 floats

  // Level 1: 256 images, tiles 32x64 -> grid (cols=4, rows=8, imgs=256)
  dim3 g1(256 / L1_TC, 256 / L1_TR, 256);
  dtcwt_level1_wmma<<<g1, 256, 0, stream>>>(x, h0o, h1o, lolo1, yhr, yhi);

  // Level 2: 256x256 -> 128x128
  dim3 g2(128 / 64, 128 / 16, 256);
  dtcwt_qshift_lo<<<g2, 256, 0, stream>>>(lolo1, lolo2, h0a, h0b, 256, 128, 128 * 128);

  // Level 3: 128x128 -> 64x64 (final Yl)
  dim3 g3(64 / 64, 64 / 16, 256);
  dtcwt_qshift_lo<<<g3, 256, 0, stream>>>(lolo2, yl, h0a, h0b, 128, 64, 64 * 64);
}